// SensoryModule_36636071034903
// MI455X (gfx1250) — compile-verified
//
#include <hip/hip_runtime.h>

// SensoryModule (muscle spindle + GTO afferents), MI455X / gfx1250.
//
// Elementwise, bandwidth-bound: 537 MB moved per call -> ~23 us floor at
// 23.3 TB/s. WMMA is intentionally NOT used: the weights form a diagonal
// scale (no K-reduction), so matrix pipes provide zero benefit. The optimal
// CDNA5 path is 128-bit coalesced global traffic with non-temporal (TH=NT)
// cache policy, since the per-pass working set exceeds the 192 MB L2 and
// every byte is touched exactly once.

typedef float v4f __attribute__((ext_vector_type(4)));

static __device__ __forceinline__ v4f nt_load(const v4f* __restrict__ p) {
    return __builtin_nontemporal_load(p);     // global_load_b128, TH=NT
}
static __device__ __forceinline__ void nt_store(v4f* __restrict__ p, v4f v) {
    __builtin_nontemporal_store(v, p);        // global_store_b128, TH=NT
}
static __device__ __forceinline__ v4f relu4(v4f a) {
    v4f r;
    r.x = fmaxf(a.x, 0.0f);
    r.y = fmaxf(a.y, 0.0f);
    r.z = fmaxf(a.z, 0.0f);
    r.w = fmaxf(a.w, 0.0f);
    return r;
}

// Each thread handles 4 consecutive muscles of one batch row.
// M = 64 muscles -> 16 threads (float4 lanes) per row; a 32-lane wave covers
// two full rows with perfectly coalesced 128B-per-8-lane segments.
__global__ __launch_bounds__(256)
void sensory_afferents_kernel(
    const v4f* __restrict__ lengths,      // [B,64] as [B*16] v4f
    const v4f* __restrict__ velocities,
    const v4f* __restrict__ forces,
    const v4f* __restrict__ gamma_static,
    const v4f* __restrict__ gamma_dynamic,
    const v4f* __restrict__ w_Ia,         // [64] as [16] v4f (cache-resident)
    const v4f* __restrict__ b_Ia,
    const v4f* __restrict__ w_II,
    const v4f* __restrict__ b_II,
    const v4f* __restrict__ w_Ib,
    const v4f* __restrict__ b_Ib,
    v4f* __restrict__ out,                // [B,192] as [B*48] v4f
    int total)                            // B*16
{
    const int tid = blockIdx.x * blockDim.x + threadIdx.x;
    if (tid >= total) return;

    const int row = tid >> 4;   // batch index
    const int m4  = tid & 15;   // which group of 4 muscles

    // Streaming inputs: non-temporal 128-bit loads (touched exactly once).
    const v4f v  = nt_load(velocities   + tid);
    const v4f gd = nt_load(gamma_dynamic+ tid);
    const v4f l  = nt_load(lengths      + tid);
    const v4f gs = nt_load(gamma_static + tid);
    const v4f f  = nt_load(forces       + tid);

    // Weights/biases: 1.5 KB total, default (RT) policy -> stays in L0/L2.
    const v4f wa = w_Ia[m4], ba = b_Ia[m4];
    const v4f wi = w_II[m4], bi = b_II[m4];
    const v4f wb = w_Ib[m4], bb = b_Ib[m4];

    const v4f type_Ia = relu4(v * gd * wa + ba);
    const v4f type_II = relu4(l * gs * wi + bi);
    const v4f type_Ib = relu4(f * wb + bb);

    // Output row = 192 floats = 48 v4f: [Ia | II | Ib]
    v4f* __restrict__ orow = out + row * 48;
    nt_store(orow + m4,      type_Ia);
    nt_store(orow + 16 + m4, type_II);
    nt_store(orow + 32 + m4, type_Ib);
}

extern "C" void kernel_launch(void* const* d_in, const int* in_sizes, int n_in,
                              void* d_out, int out_size, void* d_ws, size_t ws_size,
                              hipStream_t stream) {
    (void)n_in; (void)d_ws; (void)ws_size; (void)out_size;

    // Input order per setup_inputs():
    // 0 lengths, 1 velocities, 2 forces, 3 gamma_static, 4 gamma_dynamic,
    // 5 w_Ia, 6 b_Ia, 7 w_II, 8 b_II, 9 w_Ib, 10 b_Ib
    const v4f* lengths  = (const v4f*)d_in[0];
    const v4f* vel      = (const v4f*)d_in[1];
    const v4f* forces   = (const v4f*)d_in[2];
    const v4f* gstat    = (const v4f*)d_in[3];
    const v4f* gdyn     = (const v4f*)d_in[4];
    const v4f* w_Ia     = (const v4f*)d_in[5];
    const v4f* b_Ia     = (const v4f*)d_in[6];
    const v4f* w_II     = (const v4f*)d_in[7];
    const v4f* b_II     = (const v4f*)d_in[8];
    const v4f* w_Ib     = (const v4f*)d_in[9];
    const v4f* b_Ib     = (const v4f*)d_in[10];
    v4f* out            = (v4f*)d_out;

    const int M  = 64;
    const int B  = in_sizes[0] / M;        // 262144
    const int total = B * (M / 4);         // v4f elements per input array

    const int block = 256;                 // 8 wave32 waves
    const int grid  = (total + block - 1) / block;

    sensory_afferents_kernel<<<grid, block, 0, stream>>>(
        lengths, vel, forces, gstat, gdyn,
        w_Ia, b_Ia, w_II, b_II, w_Ib, b_Ib,
        out, total);
}